// SubMasking_68066641707079
// MI455X (gfx1250) — compile-verified
//
#include <hip/hip_runtime.h>
#include <hip/hip_bf16.h>

// SubMasking: out[b,r,c] = (r < n-1 && c > r) ? -1e9f : x[b,r,c]
// Pure streaming op. Strategy:
//   * vec4 (b128) accesses, one vec4 per thread, perfectly coalesced in wave32
//   * fully-masked vec4 chunks (c0 > r): store constant, skip the load entirely
//     (~halves read traffic since the masked region is the upper triangle)
//   * fully-kept chunks: nontemporal b128 copy
//   * the single straddling chunk per row: load + per-lane select
//   * non-temporal loads & stores (gfx1250 TH=NT): 1 GiB/call >> 192 MB L2, zero reuse

typedef float f32x4 __attribute__((ext_vector_type(4)));

#define NEGV (-1000000000.0f)
#define S_DIM 4096
#define VEC_PER_ROW (S_DIM / 4)   // 1024

__global__ __launch_bounds__(256) void SubMasking_68066641707079_kernel(
    const float* __restrict__ x,
    const int* __restrict__ n_ptr,
    float* __restrict__ out,
    unsigned int num_vec4)
{
    unsigned int i = blockIdx.x * blockDim.x + threadIdx.x;
    if (i >= num_vec4) return;

    // Flat vec4 index -> (row within matrix, starting column). vec_per_row and
    // S are powers of two, so this is shifts/ands only.
    const int r  = (int)((i >> 10) & (S_DIM - 1));  // (i / VEC_PER_ROW) % S
    const int c0 = (int)(i & (VEC_PER_ROW - 1)) << 2;

    const int n = n_ptr[0];                  // uniform scalar load
    const bool row_masked = (r < n - 1);

    const f32x4* __restrict__ xv = reinterpret_cast<const f32x4*>(x);
    f32x4* __restrict__ ov       = reinterpret_cast<f32x4*>(out);

    if (!row_masked || (c0 + 3 <= r)) {
        // Fully kept: streaming copy, NT both ways.
        f32x4 v = __builtin_nontemporal_load(&xv[i]);
        __builtin_nontemporal_store(v, &ov[i]);
    } else if (c0 > r) {
        // Fully masked: no load at all, just write the constant.
        f32x4 v;
        v.x = NEGV; v.y = NEGV; v.z = NEGV; v.w = NEGV;
        __builtin_nontemporal_store(v, &ov[i]);
    } else {
        // Straddles the diagonal (at most one vec4 per row): per-lane select.
        f32x4 v = __builtin_nontemporal_load(&xv[i]);
        v.x = (c0 + 0 > r) ? NEGV : v.x;
        v.y = (c0 + 1 > r) ? NEGV : v.y;
        v.z = (c0 + 2 > r) ? NEGV : v.z;
        v.w = (c0 + 3 > r) ? NEGV : v.w;
        __builtin_nontemporal_store(v, &ov[i]);
    }
}

extern "C" void kernel_launch(void* const* d_in, const int* in_sizes, int n_in,
                              void* d_out, int out_size, void* d_ws, size_t ws_size,
                              hipStream_t stream) {
    const float* x     = (const float*)d_in[0];
    const int*   n_ptr = (const int*)d_in[1];
    float*       out   = (float*)d_out;

    const unsigned int total    = (unsigned int)in_sizes[0];   // B*S*S = 134,217,728
    const unsigned int num_vec4 = total >> 2;                  // 33,554,432

    const int block = 256;
    const unsigned int grid = (num_vec4 + block - 1) / block;  // 131,072 blocks

    SubMasking_68066641707079_kernel<<<grid, block, 0, stream>>>(x, n_ptr, out, num_vec4);
}